// TransformerConv_36902359007745
// MI455X (gfx1250) — compile-verified
//
#include <hip/hip_runtime.h>

#define F_IN 128
#define HD   64
#define NH   4
#define EPS  1e-5f

typedef __attribute__((ext_vector_type(2))) float v2f;
typedef __attribute__((ext_vector_type(8))) float v8f;

// ---- ordered-uint encoding for float atomic max (monotone for all reals) ----
__device__ __forceinline__ unsigned f2ord(float f) {
    unsigned u = __float_as_uint(f);
    return (u & 0x80000000u) ? ~u : (u | 0x80000000u);
}
__device__ __forceinline__ float ord2f(unsigned o) {
    unsigned u = (o & 0x80000000u) ? (o & 0x7fffffffu) : ~o;
    return __uint_as_float(u);
}

// ---------------- init m (running max key) and z (softmax denom) -------------
__global__ void init_mz(unsigned* __restrict__ m_ord, float* __restrict__ z, int count) {
    int i = blockIdx.x * blockDim.x + threadIdx.x;
    if (i < count) { m_ord[i] = 0u; z[i] = 0.0f; }   // key 0 == "-inf"
}

// ---------------- fused q/k/v/skip projections via f32 WMMA ------------------
// 8 waves per block; block = 16 node rows. Wave w: matrix m = w>>1,
// column half = (w&1)*32 (two 16x16 N-tiles). K loop: 32 steps of 16x16x4.
__global__ __launch_bounds__(256) void qkv_gemm(
    const float* __restrict__ feat,
    const float* __restrict__ Wq, const float* __restrict__ bq,
    const float* __restrict__ Wk, const float* __restrict__ bk,
    const float* __restrict__ Wv, const float* __restrict__ bv,
    const float* __restrict__ Wsk, const float* __restrict__ bsk,
    float* __restrict__ q, float* __restrict__ k, float* __restrict__ v,
    float* __restrict__ accum, int n)
{
    __shared__ float ftile[16 * 132];             // pad 128->132: no bank conflicts
    const int tid = threadIdx.x;
    const int blockRow = blockIdx.x * 16;

    // stage 16x128 feature tile (2048 floats, 8 per thread)
    for (int i = 0; i < 8; ++i) {
        int idx = tid + i * 256;
        int r = idx >> 7, c = idx & 127;
        int gr = blockRow + r;
        ftile[r * 132 + c] = (gr < n) ? feat[(size_t)gr * F_IN + c] : 0.0f;
    }
    __syncthreads();

    const int wave = tid >> 5;
    const int lane = tid & 31;
    const int m    = wave >> 1;
    const int n0   = (wave & 1) * 32;
    const int hv   = lane >> 4;                   // 0: lanes 0-15, 1: lanes 16-31
    const int l16  = lane & 15;

    const float* W = (m == 0) ? Wq : (m == 1) ? Wk : (m == 2) ? Wv : Wsk;
    const float* B = (m == 0) ? bq : (m == 1) ? bk : (m == 2) ? bv : bsk;
    float*       O = (m == 0) ? q  : (m == 1) ? k  : (m == 2) ? v  : accum;

    v8f c0 = {}; v8f c1 = {};
    const float* arow = &ftile[l16 * 132];
    for (int kk = 0; kk < F_IN / 4; ++kk) {
        const int k0 = kk * 4 + 2 * hv;           // A: lanes0-15 K={4kk,4kk+1}, lanes16-31 K={4kk+2,4kk+3}
        v2f a; a.x = arow[k0]; a.y = arow[k0 + 1];
        const float* w0 = W + (size_t)k0 * HD + n0 + l16;
        v2f b0; b0.x = w0[0];  b0.y = w0[HD];
        v2f b1; b1.x = w0[16]; b1.y = w0[HD + 16];
        c0 = __builtin_amdgcn_wmma_f32_16x16x4_f32(false, a, false, b0, (short)0, c0, false, false);
        c1 = __builtin_amdgcn_wmma_f32_16x16x4_f32(false, a, false, b1, (short)0, c1, false, false);
    }

    const float scale = (m == 0) ? 0.25f : 1.0f;  // q *= 1/sqrt(D)
    const int col0 = n0 + l16;
    for (int r = 0; r < 8; ++r) {                 // C: vgpr r -> M = r + 8*hv, N = lane%16
        int row = blockRow + r + 8 * hv;
        if (row < n) {
            O[(size_t)row * HD + col0]      = (c0[r] + B[col0])      * scale;
            O[(size_t)row * HD + col0 + 16] = (c1[r] + B[col0 + 16]) * scale;
        }
    }
}

// ---------------- edge scores + segment max ---------------------------------
__global__ void edge_alpha(const float* __restrict__ q, const float* __restrict__ k,
                           const int* __restrict__ src, const int* __restrict__ dst,
                           float* __restrict__ alpha, unsigned* __restrict__ m_ord, int ecount)
{
    int e = blockIdx.x * blockDim.x + threadIdx.x;
    if (e >= ecount) return;
    int s = src[e], d = dst[e];
    const float4* qd = (const float4*)(q + (size_t)d * HD);
    const float4* ks = (const float4*)(k + (size_t)s * HD);
    #pragma unroll
    for (int h = 0; h < NH; ++h) {
        float acc = 0.0f;
        #pragma unroll
        for (int j = 0; j < 4; ++j) {
            float4 a = qd[h * 4 + j], b = ks[h * 4 + j];
            acc += a.x * b.x + a.y * b.y + a.z * b.z + a.w * b.w;
        }
        alpha[(size_t)e * NH + h] = acc;
        atomicMax(&m_ord[(size_t)d * NH + h], f2ord(acc));
    }
}

// ---------------- exp(alpha - m), segment sum into z -------------------------
__global__ void edge_exp(const int* __restrict__ dst, const unsigned* __restrict__ m_ord,
                         float* __restrict__ alpha, float* __restrict__ z, int count)
{
    int i = blockIdx.x * blockDim.x + threadIdx.x;
    if (i >= count) return;
    int e = i >> 2, h = i & 3;
    int d = dst[e];
    float mm = ord2f(m_ord[(size_t)d * NH + h]);
    float ex = __expf(alpha[i] - mm);
    alpha[i] = ex;                                 // overwrite with numerator
    atomicAdd(&z[(size_t)d * NH + h], ex);
}

// ---------------- weighted message scatter-add -------------------------------
// 16 threads/edge; thread owns 4 consecutive channels (one head chunk).
__global__ void edge_msg(const int* __restrict__ src, const int* __restrict__ dst,
                         const float* __restrict__ v, const float* __restrict__ alpha,
                         const float* __restrict__ z, float* __restrict__ accum, int count)
{
    int i = blockIdx.x * blockDim.x + threadIdx.x;
    if (i >= count) return;
    int e = i >> 4, part = i & 15;
    int h = part >> 2;
    int s = src[e], d = dst[e];
    float a = alpha[(size_t)e * NH + h] / z[(size_t)d * NH + h];
    float4 vv = ((const float4*)(v + (size_t)s * HD))[part];
    float* ap = accum + (size_t)d * HD + part * 4;
    atomicAdd(ap + 0, vv.x * a);
    atomicAdd(ap + 1, vv.y * a);
    atomicAdd(ap + 2, vv.z * a);
    atomicAdd(ap + 3, vv.w * a);
}

// ---------------- LayerNorm: one wave32 per node -----------------------------
__global__ __launch_bounds__(256) void ln_kernel(const float* __restrict__ accum,
                                                 const float* __restrict__ g,
                                                 const float* __restrict__ b,
                                                 float* __restrict__ out, int n)
{
    int wave = threadIdx.x >> 5;
    int lane = threadIdx.x & 31;
    int node = blockIdx.x * 8 + wave;
    if (node >= n) return;
    float2 x = ((const float2*)(accum + (size_t)node * HD))[lane];
    float s = x.x + x.y;
    for (int msk = 16; msk > 0; msk >>= 1) s += __shfl_xor(s, msk, 32);
    float mean = s * (1.0f / HD);
    float e0 = x.x - mean, e1 = x.y - mean;
    float s2 = e0 * e0 + e1 * e1;
    for (int msk = 16; msk > 0; msk >>= 1) s2 += __shfl_xor(s2, msk, 32);
    float rstd = rsqrtf(s2 * (1.0f / HD) + EPS);
    int c = lane * 2;
    float2 o;
    o.x = e0 * rstd * g[c]     + b[c];
    o.y = e1 * rstd * g[c + 1] + b[c + 1];
    ((float2*)(out + (size_t)node * HD))[lane] = o;
}

extern "C" void kernel_launch(void* const* d_in, const int* in_sizes, int n_in,
                              void* d_out, int out_size, void* d_ws, size_t ws_size,
                              hipStream_t stream)
{
    const float* feat = (const float*)d_in[0];
    const int*   src  = (const int*)  d_in[1];
    const int*   dst  = (const int*)  d_in[2];
    const float* Wq   = (const float*)d_in[3];
    const float* bq   = (const float*)d_in[4];
    const float* Wk   = (const float*)d_in[5];
    const float* bk   = (const float*)d_in[6];
    const float* Wv   = (const float*)d_in[7];
    const float* bv   = (const float*)d_in[8];
    const float* Wsk  = (const float*)d_in[9];
    const float* bsk  = (const float*)d_in[10];
    const float* ln_g = (const float*)d_in[11];
    const float* ln_b = (const float*)d_in[12];

    const int n = in_sizes[0] / F_IN;   // 100000
    const int e = in_sizes[1];          // 1600000

    char* ws = (char*)d_ws;
    size_t off = 0;
    float*    q     = (float*)(ws + off); off += (size_t)n * HD * 4;
    float*    k     = (float*)(ws + off); off += (size_t)n * HD * 4;
    float*    v     = (float*)(ws + off); off += (size_t)n * HD * 4;
    float*    accum = (float*)(ws + off); off += (size_t)n * HD * 4;   // init'd by skip write
    float*    alpha = (float*)(ws + off); off += (size_t)e * NH * 4;
    float*    z     = (float*)(ws + off); off += (size_t)n * NH * 4;
    unsigned* m_ord = (unsigned*)(ws + off); off += (size_t)n * NH * 4;

    init_mz<<<(n * NH + 255) / 256, 256, 0, stream>>>(m_ord, z, n * NH);
    qkv_gemm<<<(n + 15) / 16, 256, 0, stream>>>(feat, Wq, bq, Wk, bk, Wv, bv, Wsk, bsk,
                                                q, k, v, accum, n);
    edge_alpha<<<(e + 255) / 256, 256, 0, stream>>>(q, k, src, dst, alpha, m_ord, e);
    edge_exp<<<(e * NH + 255) / 256, 256, 0, stream>>>(dst, m_ord, alpha, z, e * NH);
    edge_msg<<<(e * 16 + 255) / 256, 256, 0, stream>>>(src, dst, v, alpha, z, accum, e * 16);
    ln_kernel<<<(n + 7) / 8, 256, 0, stream>>>(accum, ln_g, ln_b, (float*)d_out, n);
}